// GraphFlow_64484638982603
// MI455X (gfx1250) — compile-verified
//
#include <hip/hip_runtime.h>
#include <cstdint>

// GraphFlow LU-flow: z <- ((z @ L_k) * d_k) @ U_k + b_k, k = 0..3, D = 3.
// out[e] = {z0, z1, z2, sum_{k,d} log|0.1 + w_kdd|}
//
// Roofline: 204 B in + 16 B out per element, B = 2^21 => ~462 MB total at
// 23.3 TB/s => ~20 us; ~90 FLOPs/elem => memory-bound by >100x. Per-element
// 3x3 LU weights share no operand across elements => WMMA inapplicable.
// CDNA5 path used instead: async global->LDS staging (ASYNCcnt) giving fully
// contiguous 16B/lane bursts, then conflict-free ds_load_b128 re-reads
// (element stride 9 chunks, coprime to the 16 super-banks).

#define FLOW_EPS 0.1f

constexpr int BLK = 256;  // elements (== threads) per block

// Core per-element math. w36: 4 chunks of 9 floats (stride 9), b12: 4x3.
__device__ __forceinline__ float4 lu_flow_elem(float z0, float z1, float z2,
                                               const float* __restrict__ w36,
                                               const float* __restrict__ b12) {
  float logdet = 0.0f;
#pragma unroll
  for (int k = 0; k < 4; ++k) {
    const float* wk = w36 + k * 9;
    const float w00 = wk[0], w01 = wk[1], w02 = wk[2];
    const float w10 = wk[3], w11 = wk[4], w12 = wk[5];
    const float w20 = wk[6], w21 = wk[7], w22 = wk[8];
    const float b0 = b12[k * 3 + 0], b1 = b12[k * 3 + 1], b2 = b12[k * 3 + 2];

    const float d0 = fabsf(w00) + FLOW_EPS;
    const float d1 = fabsf(w11) + FLOW_EPS;
    const float d2 = fabsf(w22) + FLOW_EPS;

    // y = z @ L   (unit lower-triangular)
    const float y0 = fmaf(z1, w10, fmaf(z2, w20, z0));
    const float y1 = fmaf(z2, w21, z1);
    const float y2 = z2;
    // t = y * diag(d)
    const float t0 = y0 * d0, t1 = y1 * d1, t2 = y2 * d2;
    // z' = t @ U + b   (unit upper-triangular)
    z0 = t0 + b0;
    z1 = fmaf(t0, w01, t1) + b1;
    z2 = fmaf(t0, w02, fmaf(t1, w12, t2)) + b2;

    // reference: sum log|EPS + diag(w)|  (abs AFTER adding EPS)
    logdet += logf(fabsf(FLOW_EPS + w00) * fabsf(FLOW_EPS + w11) *
                   fabsf(FLOW_EPS + w22));
  }
  return make_float4(z0, z1, z2, logdet);
}

__device__ __forceinline__ void async_copy_b128(uint32_t lds_addr,
                                                uint32_t goffset,
                                                uint64_t gbase) {
  // vdst = LDS byte address (vgpr), vaddr = 32-bit byte offset (vgpr),
  // saddr = 64-bit global base (sgpr pair). Tracked by ASYNCcnt.
  asm volatile("global_load_async_to_lds_b128 %0, %1, %2"
               :
               : "v"(lds_addr), "v"(goffset), "s"(gbase)
               : "memory");
}

__global__ __launch_bounds__(BLK) void graphflow_lu_kernel(
    const float* __restrict__ z, const float* __restrict__ w,
    const float* __restrict__ b, float* __restrict__ out, int B) {
  __shared__ float sw[BLK * 36];  // 36864 B, 144 B (9 x 16B chunks) / element
  __shared__ float sb[BLK * 12];  // 12288 B, 48 B (3 chunks) / element
  __shared__ float sz[BLK * 3];   //  3072 B
  // Element stride in 16B chunks: w=9, b=3 -- both coprime to the 16
  // super-banks => ds_load_b128 re-reads are bank-conflict-free.

  const int tid = threadIdx.x;
  const int e0  = blockIdx.x * BLK;
  const bool full_block = (e0 + BLK) <= B;  // uniform per block

  if (full_block) {
    // Flat pointers to LDS carry the LDS byte offset in their low 32 bits
    // (LDS aperture base has zero low bits), so truncation gives the DS addr.
    const uint32_t sw_base = (uint32_t)(uintptr_t)&sw[0];
    const uint32_t sb_base = (uint32_t)(uintptr_t)&sb[0];
    const uint32_t sz_base = (uint32_t)(uintptr_t)&sz[0];

    // All three copies are plain contiguous block transfers: lane chunk id
    // c maps to byte offset c*16 on BOTH sides -- no index math, contiguous
    // 512 B global bursts per wave32 instruction, contiguous LDS DMA writes.
    {
      const uint64_t gbase = (uint64_t)(uintptr_t)(w + (size_t)e0 * 36);
#pragma unroll
      for (int i = 0; i < 9; ++i) {  // 2304 chunks of 16 B
        const uint32_t off = (uint32_t)(tid + i * BLK) * 16u;
        async_copy_b128(sw_base + off, off, gbase);
      }
    }
    {
      const uint64_t gbase = (uint64_t)(uintptr_t)(b + (size_t)e0 * 12);
#pragma unroll
      for (int i = 0; i < 3; ++i) {  // 768 chunks
        const uint32_t off = (uint32_t)(tid + i * BLK) * 16u;
        async_copy_b128(sb_base + off, off, gbase);
      }
    }
    {
      const uint64_t gbase = (uint64_t)(uintptr_t)(z + (size_t)e0 * 3);
      if (tid < 192) {  // 192 chunks
        const uint32_t off = (uint32_t)tid * 16u;
        async_copy_b128(sz_base + off, off, gbase);
      }
    }

    asm volatile("s_wait_asynccnt 0x0" ::: "memory");
    __syncthreads();

    // Deterministic ds_load_b128 re-reads into registers.
    float wl[36];
    {
      const float4* wp4 = reinterpret_cast<const float4*>(sw + tid * 36);
#pragma unroll
      for (int j = 0; j < 9; ++j)
        reinterpret_cast<float4*>(wl)[j] = wp4[j];
    }
    float bl[12];
    {
      const float4* bp4 = reinterpret_cast<const float4*>(sb + tid * 12);
#pragma unroll
      for (int j = 0; j < 3; ++j)
        reinterpret_cast<float4*>(bl)[j] = bp4[j];
    }
    const float z0 = sz[tid * 3 + 0];
    const float z1 = sz[tid * 3 + 1];
    const float z2 = sz[tid * 3 + 2];

    const float4 o = lu_flow_elem(z0, z1, z2, wl, bl);
    *reinterpret_cast<float4*>(out + (size_t)(e0 + tid) * 4) = o;
  } else {
    // Tail path (B % 256 != 0): direct global loads, no LDS, no barrier.
    const int e = e0 + tid;
    if (e < B) {
      const float* wp = w + (size_t)e * 36;
      const float* bp = b + (size_t)e * 12;
      const float4 o  = lu_flow_elem(z[(size_t)e * 3 + 0], z[(size_t)e * 3 + 1],
                                     z[(size_t)e * 3 + 2], wp, bp);
      *reinterpret_cast<float4*>(out + (size_t)e * 4) = o;
    }
  }
}

extern "C" void kernel_launch(void* const* d_in, const int* in_sizes, int n_in,
                              void* d_out, int out_size, void* d_ws,
                              size_t ws_size, hipStream_t stream) {
  const float* z = (const float*)d_in[0];  // [B, 3]
  const float* w = (const float*)d_in[1];  // [B, 4, 3, 3]
  const float* b = (const float*)d_in[2];  // [B, 4, 3]
  float* out     = (float*)d_out;          // [B, 4]

  const int B    = in_sizes[0] / 3;
  const int grid = (B + BLK - 1) / BLK;
  graphflow_lu_kernel<<<grid, BLK, 0, stream>>>(z, w, b, out, B);
}